// MedianFilter3DDepthwise_85126251807212
// MI455X (gfx1250) — compile-verified
//
#include <hip/hip_runtime.h>
#include <stdint.h>

// ---------------------------------------------------------------------------
// 5x5 exact median filter, per depth slice, REFLECT padding.
// x: [1, 32, 256, 256, 8] f32  ->  out same shape.
//
// MI455X reasoning:
//  * Memory floor: 64MB in + 64MB out => ~5.5us @ 23.3 TB/s. Input fits in
//    192MB L2, but we still tile through LDS (async-to-LDS, ASYNCcnt) and
//    slide the 5x5 window vertically in registers so LDS traffic is ~5 reads
//    per output instead of 25.
//  * Compute: exact median-of-25 via forgetful selection (start 14, pairwise
//    min/max stages, final v_med3_f32) ~129 compare-exchanges per output.
//  * wave32: lanes map to (w,c) with c innermost -> 128B coalesced segments
//    for loads and stores.
// ---------------------------------------------------------------------------

#define DDIM 32
#define HDIM 256
#define WDIM 256
#define CH   8

#define TH 8                 // output rows per block
#define TW 32                // output w-pixels per block
#define HALO_R (TH + 4)      // 12
#define HALO_W (TW + 4)      // 36
#define HALO_PIX (HALO_R * HALO_W)      // 432 pixels; 2 x b128 chunks each
#define HALO_CHUNKS (HALO_PIX * 2)      // 864

__device__ __forceinline__ void ce(float& x, float& y) {
    float lo = fminf(x, y);
    float hi = fmaxf(x, y);
    x = lo; y = hi;
}

__device__ __forceinline__ float med3(float a, float b, float c) {
#if __has_builtin(__builtin_amdgcn_fmed3f)
    return __builtin_amdgcn_fmed3f(a, b, c);
#else
    return fmaxf(fminf(a, b), fminf(fmaxf(a, b), c));
#endif
}

// np.pad mode='reflect' (mirror without repeating the edge), pad=2, n>=3
__device__ __forceinline__ int reflect_idx(int i, int n) {
    if (i < 0) i = -i;
    if (i >= n) i = 2 * n - 2 - i;
    return i;
}

// Exact median of 25 via forgetful selection. The median survives every
// stage: the min (max) of each working set is transitively <= (>=) at least
// 13 of the 25 values, so it cannot be the rank-13 element.
__device__ __forceinline__ float median25(const float v[25]) {
    float a[14];
#pragma unroll
    for (int i = 0; i < 14; ++i) a[i] = v[i];
    int next = 14;
#pragma unroll
    for (int s = 14; s >= 4; --s) {
        // pairwise: min ends in low half, max in high half
#pragma unroll
        for (int i = 0; i < s / 2; ++i) ce(a[i], a[s - 1 - i]);
        // min-reduce low half into a[0]
#pragma unroll
        for (int i = 1; i < (s + 1) / 2; ++i) ce(a[0], a[i]);
        // max-reduce high half into a[s-1]
#pragma unroll
        for (int i = s / 2; i < s - 1; ++i) ce(a[i], a[s - 1]);
        // discard a[0] (min) and a[s-1] (max); shift; append next value
#pragma unroll
        for (int j = 0; j < s - 2; ++j) a[j] = a[j + 1];
        a[s - 2] = v[next++];
    }
    return med3(a[0], a[1], a[2]);
}

__global__ __launch_bounds__(256)
void median5x5_kernel(const float* __restrict__ x, float* __restrict__ out) {
    __shared__ __align__(16) float tile[HALO_R][HALO_W][CH];

    const int tid = threadIdx.x;
    const int w0  = blockIdx.x * TW;
    const int h0  = blockIdx.y * TH;
    const int d   = blockIdx.z;

    // ---- stage halo tile into LDS with async copies (ASYNCcnt path) ----
    const unsigned lds_base = (unsigned)(uintptr_t)(&tile[0][0][0]);
    const uint64_t gbase =
        (uint64_t)(uintptr_t)x + (uint64_t)d * (HDIM * (uint64_t)WDIM * CH * 4);

#pragma unroll
    for (int k = 0; k < 4; ++k) {
        int q = k * 256 + tid;
        if (q < HALO_CHUNKS) {
            int half = q / HALO_PIX;              // 0: channels 0..3, 1: 4..7
            int pix  = q - half * HALO_PIX;
            int pr   = pix / HALO_W;
            int pw   = pix - pr * HALO_W;
            int gh   = reflect_idx(h0 + pr - 2, HDIM);
            int gw   = reflect_idx(w0 + pw - 2, WDIM);
            uint64_t gaddr = gbase +
                (uint64_t)(((gh * WDIM + gw) * CH + half * 4) * 4);
            unsigned laddr = lds_base +
                (unsigned)(((pr * HALO_W + pw) * CH + half * 4) * 4);
            // GLOBAL_LOAD_ASYNC_TO_LDS_B128: vdst = LDS byte addr (VGPR),
            // vaddr = 64-bit global addr, saddr = off (GV mode).
            asm volatile("global_load_async_to_lds_b128 %0, %1, off"
                         :: "v"(laddr), "v"(gaddr)
                         : "memory");
        }
    }
    asm volatile("s_wait_asynccnt 0x0" ::: "memory");
    __syncthreads();

    // ---- compute: one (w,c) column per thread, slide 8 rows vertically ----
    const int ci = tid & (CH - 1);     // 0..7  (innermost -> coalesced)
    const int wi = tid >> 3;           // 0..31

    float win[5][5];                   // win[dx][dy]
#pragma unroll
    for (int j = 0; j < 5; ++j)
#pragma unroll
        for (int i = 0; i < 5; ++i)
            win[j][i] = tile[i][wi + j][ci];

#pragma unroll
    for (int r = 0; r < TH; ++r) {
        float v[25];
#pragma unroll
        for (int j = 0; j < 5; ++j)
#pragma unroll
            for (int i = 0; i < 5; ++i)
                v[j * 5 + i] = win[j][i];

        float m = median25(v);

        size_t oidx = ((size_t)((d * HDIM + h0 + r) * WDIM + (w0 + wi))) * CH + ci;
        out[oidx] = m;

        if (r < TH - 1) {
#pragma unroll
            for (int j = 0; j < 5; ++j) {
#pragma unroll
                for (int i = 0; i < 4; ++i) win[j][i] = win[j][i + 1];
                win[j][4] = tile[r + 5][wi + j][ci];
            }
        }
    }
}

extern "C" void kernel_launch(void* const* d_in, const int* in_sizes, int n_in,
                              void* d_out, int out_size, void* d_ws, size_t ws_size,
                              hipStream_t stream) {
    (void)in_sizes; (void)n_in; (void)d_ws; (void)ws_size; (void)out_size;
    const float* x = (const float*)d_in[0];
    float* out = (float*)d_out;
    dim3 grid(WDIM / TW, HDIM / TH, DDIM);   // 8 x 32 x 32 = 8192 blocks
    median5x5_kernel<<<grid, 256, 0, stream>>>(x, out);
}